// TransformerBlockWithCache_16544214024833
// MI455X (gfx1250) — compile-verified
//
#include <hip/hip_runtime.h>
#include <hip/hip_bf16.h>
#include <math.h>

#define E_DIM 1024
#define S_LEN 8192
#define H_N   16
#define D_H   64

typedef __attribute__((ext_vector_type(2))) float v2f;
typedef __attribute__((ext_vector_type(8))) float v8f;

// D = A(16x4 f32) * B(4x16 f32) + C(16x16 f32), wave32 collective.
__device__ __forceinline__ v8f wmma4(v2f a, v2f b, v8f c) {
    return __builtin_amdgcn_wmma_f32_16x16x4_f32(false, a, false, b, (short)0, c,
                                                 false, false);
}

// ---------------------------------------------------------------------------
// Kernel 1: qkv partial GEMVs. grid = (8 out-groups, 4 k-slices, 3 matrices).
// Each wave -> 16 outputs over a 256-row K slice; partials to ws (no atomics).
// ---------------------------------------------------------------------------
__global__ void qkv_part_kernel(const float* __restrict__ x,
                                const float* __restrict__ Wv,
                                const float* __restrict__ Wq,
                                const float* __restrict__ Wk,
                                float* __restrict__ part_qkv) {
    __shared__ float xs[256];
    const int tid = threadIdx.x;
    const int k0  = blockIdx.y * 256;
    if (tid < 256) xs[tid] = x[k0 + tid];
    __syncthreads();

    const int m    = blockIdx.z;
    const int wv   = tid >> 5;
    const int lane = tid & 31;
    const int j    = lane & 15;
    const int half = lane >> 4;
    const int ob   = blockIdx.x * 128 + wv * 16;
    const float amask = (j == 0) ? 1.f : 0.f;

    const float* W = (m == 0) ? Wv : (m == 1) ? Wq : Wk;

    v8f c0 = {}, c1 = {};
    for (int eb = 0; eb < 256; eb += 8) {
        const int r0 = eb + half * 2;
        v2f a0, a1, b0, b1;
        a0.x = xs[r0]     * amask;  a0.y = xs[r0 + 1] * amask;
        a1.x = xs[r0 + 4] * amask;  a1.y = xs[r0 + 5] * amask;
        const float* wp0 = W + (size_t)(k0 + r0)     * E_DIM + ob + j;
        const float* wp1 = W + (size_t)(k0 + r0 + 4) * E_DIM + ob + j;
        b0.x = wp0[0];  b0.y = wp0[E_DIM];
        b1.x = wp1[0];  b1.y = wp1[E_DIM];
        c0 = wmma4(a0, b0, c0);
        c1 = wmma4(a1, b1, c1);
    }
    const v8f cs = c0 + c1;
    if (lane < 16)
        part_qkv[(size_t)(m * 4 + blockIdx.y) * E_DIM + ob + lane] = cs[0];
}

// ---------------------------------------------------------------------------
// Kernel 2: combine qkv partials (+bias); write q to ws, v_i/k_i into cache row.
// ---------------------------------------------------------------------------
__global__ void qkv_combine_kernel(const float* __restrict__ part_qkv,
                                   const float* __restrict__ bv,
                                   const float* __restrict__ bq,
                                   const float* __restrict__ bk,
                                   float* __restrict__ q_out,
                                   float* __restrict__ newv, float* __restrict__ newk) {
    for (int o = threadIdx.x; o < 3 * E_DIM; o += 256) {
        const int m  = o >> 10;
        const int oo = o & (E_DIM - 1);
        float s = 0.f;
        #pragma unroll
        for (int ks = 0; ks < 4; ks++)
            s += part_qkv[(size_t)(m * 4 + ks) * E_DIM + oo];
        if (m == 0)      newv[(size_t)(S_LEN - 1) * E_DIM + oo] = s + bv[oo];
        else if (m == 1) q_out[oo] = s + bq[oo];
        else             newk[(size_t)(S_LEN - 1) * E_DIM + oo] = s + bk[oo];
    }
}

// ---------------------------------------------------------------------------
// Kernel 3: fused cache shift-copy + mask + q.k scores. 256 blocks x 256.
// ---------------------------------------------------------------------------
__global__ void shift_score_kernel(const float* __restrict__ oldv,
                                   const float* __restrict__ oldk,
                                   const float* __restrict__ q,
                                   float* __restrict__ newv, float* __restrict__ newk,
                                   float* __restrict__ scores) {
    const int tid  = threadIdx.x;
    const int lane = tid & 31;
    const int wv   = tid >> 5;
    const int c0   = blockIdx.x * 32;
    const float4* q4 = (const float4*)q;

    for (int p = wv; p < 32; p += 8) {
        const int s = c0 + p;
        const bool last = (s == S_LEN - 1);   // wave-uniform

        const float4* vsrc = (const float4*)(last ? (newv + (size_t)s * E_DIM)
                                                  : (oldv + (size_t)(s + 1) * E_DIM));
        float4* vdst = (float4*)(newv + (size_t)s * E_DIM);
        int nz = 0;
        #pragma unroll
        for (int i = 0; i < 8; i++) {
            float4 d = vsrc[i * 32 + lane];
            nz |= (d.x != 0.f) | (d.y != 0.f) | (d.z != 0.f) | (d.w != 0.f);
            if (!last) vdst[i * 32 + lane] = d;
        }
        #pragma unroll
        for (int mm = 16; mm >= 1; mm >>= 1) nz |= __shfl_xor(nz, mm, 32);

        const float4* ksrc = (const float4*)(last ? (newk + (size_t)s * E_DIM)
                                                  : (oldk + (size_t)(s + 1) * E_DIM));
        float4* kdst = (float4*)(newk + (size_t)s * E_DIM);
        #pragma unroll
        for (int i = 0; i < 8; i++) {
            float4 kd = ksrc[i * 32 + lane];
            if (!last) kdst[i * 32 + lane] = kd;
            float4 qd = q4[i * 32 + lane];
            float part = kd.x * qd.x + kd.y * qd.y + kd.z * qd.z + kd.w * qd.w;
            part += __shfl_xor(part, 1, 32);
            part += __shfl_xor(part, 2, 32);
            part += __shfl_xor(part, 4, 32);
            part += __shfl_xor(part, 8, 32);
            if ((lane & 15) == 0) {
                const int h = i * 2 + (lane >> 4);
                scores[(size_t)h * S_LEN + s] = nz ? part * 0.125f : -INFINITY;
            }
        }
    }
}

// ---------------------------------------------------------------------------
// Kernel 4: softmax per head (in-place) + zero the attn.V accumulators.
// ---------------------------------------------------------------------------
__global__ void softmax_kernel(float* __restrict__ scores, float* __restrict__ values) {
    __shared__ float red[256];
    const int h = blockIdx.x, tid = threadIdx.x;
    float* row = scores + (size_t)h * S_LEN;
    if (tid < D_H) values[h * D_H + tid] = 0.f;

    float mx = -INFINITY;
    for (int i = tid; i < S_LEN; i += 256) mx = fmaxf(mx, row[i]);
    red[tid] = mx; __syncthreads();
    for (int st = 128; st > 0; st >>= 1) {
        if (tid < st) red[tid] = fmaxf(red[tid], red[tid + st]);
        __syncthreads();
    }
    mx = red[0]; __syncthreads();

    float sm = 0.f;
    for (int i = tid; i < S_LEN; i += 256) sm += expf(row[i] - mx);
    red[tid] = sm; __syncthreads();
    for (int st = 128; st > 0; st >>= 1) {
        if (tid < st) red[tid] += red[tid + st];
        __syncthreads();
    }
    const float inv = 1.f / red[0];
    for (int i = tid; i < S_LEN; i += 256) row[i] = expf(row[i] - mx) * inv;
}

// ---------------------------------------------------------------------------
// Kernel 5: values[h,d] = sum_s p[h,s] * Vnew[s,h,d] via WMMA. 128 blocks.
// ---------------------------------------------------------------------------
__global__ void attnv_kernel(const float* __restrict__ probs,
                             const float* __restrict__ newv,
                             float* __restrict__ values) {
    const int tid  = threadIdx.x;
    const int lane = tid & 31;
    const int wv   = tid >> 5;
    const int gw   = blockIdx.x * 8 + wv;   // 0..1023
    const int h    = gw >> 6;
    const int dt   = (gw >> 4) & 3;
    const int sc   = gw & 15;
    const int s0   = sc * 512;
    const int j    = lane & 15;
    const int half = lane >> 4;
    const float amask = (j == 0) ? 1.f : 0.f;

    const float* p = probs + (size_t)h * S_LEN;
    const float* V = newv + h * D_H + dt * 16 + j;

    v8f c0 = {}, c1 = {};
    for (int sb = 0; sb < 512; sb += 8) {
        const int sk0 = s0 + sb + half * 2;
        const int sk1 = sk0 + 4;
        v2f a0, a1, b0, b1;
        a0.x = p[sk0]     * amask;  a0.y = p[sk0 + 1] * amask;
        a1.x = p[sk1]     * amask;  a1.y = p[sk1 + 1] * amask;
        const float* vp0 = V + (size_t)sk0 * E_DIM;
        const float* vp1 = V + (size_t)sk1 * E_DIM;
        b0.x = vp0[0];  b0.y = vp0[E_DIM];
        b1.x = vp1[0];  b1.y = vp1[E_DIM];
        c0 = wmma4(a0, b0, c0);
        c1 = wmma4(a1, b1, c1);
    }
    const v8f cs = c0 + c1;
    if (lane < 16) atomicAdd(&values[h * D_H + dt * 16 + lane], cs[0]);
}

// ---------------------------------------------------------------------------
// Kernel 6: K-sliced 1024x1024 GEMV partials (WMMA).
// grid = (8 out-groups, KS k-slices). Input vector can itself be a sum of P
// partials (+bias, +relu) materialized in the LDS preload.
// ---------------------------------------------------------------------------
__global__ void gemv_part_kernel(const float* __restrict__ W,
                                 const float* __restrict__ in_parts, int P,
                                 const float* __restrict__ in_bias, int in_relu,
                                 float* __restrict__ out_parts) {
    __shared__ float xs[128];
    const int tid = threadIdx.x;
    const int k0  = blockIdx.y * 128;
    if (tid < 128) {
        float s = 0.f;
        for (int pp = 0; pp < P; pp++) s += in_parts[(size_t)pp * E_DIM + k0 + tid];
        if (in_bias) s += in_bias[k0 + tid];
        if (in_relu) s = fmaxf(s, 0.f);
        xs[tid] = s;
    }
    __syncthreads();

    const int wv   = tid >> 5;
    const int lane = tid & 31;
    const int j    = lane & 15;
    const int half = lane >> 4;
    const int ob   = blockIdx.x * 128 + wv * 16;
    const float amask = (j == 0) ? 1.f : 0.f;

    v8f c0 = {}, c1 = {};
    for (int eb = 0; eb < 128; eb += 8) {
        const int r0 = eb + half * 2;
        v2f a0, a1, b0, b1;
        a0.x = xs[r0]     * amask;  a0.y = xs[r0 + 1] * amask;
        a1.x = xs[r0 + 4] * amask;  a1.y = xs[r0 + 5] * amask;
        const float* wp0 = W + (size_t)(k0 + r0)     * E_DIM + ob + j;
        const float* wp1 = W + (size_t)(k0 + r0 + 4) * E_DIM + ob + j;
        b0.x = wp0[0];  b0.y = wp0[E_DIM];
        b1.x = wp1[0];  b1.y = wp1[E_DIM];
        c0 = wmma4(a0, b0, c0);
        c1 = wmma4(a1, b1, c1);
    }
    const v8f cs = c0 + c1;
    if (lane < 16)
        out_parts[(size_t)blockIdx.y * E_DIM + ob + lane] = cs[0];
}

// ---------------------------------------------------------------------------
// Kernel 7: combine GEMV partials (+bias +residual) then LayerNorm. 1 block.
// ---------------------------------------------------------------------------
__global__ void ln_combine_kernel(const float* __restrict__ parts, int P,
                                  const float* __restrict__ bias,
                                  const float* __restrict__ resid,
                                  const float* __restrict__ sc,
                                  const float* __restrict__ bs,
                                  float* __restrict__ out) {
    __shared__ float red[256];
    const int tid = threadIdx.x;
    float vv[4];
    float sum = 0.f;
    #pragma unroll
    for (int i = 0; i < 4; i++) {
        const int e = i * 256 + tid;
        float s = bias[e] + resid[e];
        for (int pp = 0; pp < P; pp++) s += parts[(size_t)pp * E_DIM + e];
        vv[i] = s;
        sum += s;
    }
    red[tid] = sum; __syncthreads();
    for (int st = 128; st > 0; st >>= 1) {
        if (tid < st) red[tid] += red[tid + st];
        __syncthreads();
    }
    const float mean = red[0] * (1.f / E_DIM); __syncthreads();
    float var = 0.f;
    #pragma unroll
    for (int i = 0; i < 4; i++) { const float d = vv[i] - mean; var += d * d; }
    red[tid] = var; __syncthreads();
    for (int st = 128; st > 0; st >>= 1) {
        if (tid < st) red[tid] += red[tid + st];
        __syncthreads();
    }
    const float inv = rsqrtf(red[0] * (1.f / E_DIM) + 1e-6f);
    #pragma unroll
    for (int i = 0; i < 4; i++) {
        const int e = i * 256 + tid;
        out[e] = (vv[i] - mean) * inv * sc[e] + bs[e];
    }
}

extern "C" void kernel_launch(void* const* d_in, const int* in_sizes, int n_in,
                              void* d_out, int out_size, void* d_ws, size_t ws_size,
                              hipStream_t stream) {
    (void)in_sizes; (void)n_in; (void)out_size; (void)ws_size;
    const float* x     = (const float*)d_in[0];
    const float* cache = (const float*)d_in[1];
    const float* Wv = (const float*)d_in[2];  const float* bv = (const float*)d_in[3];
    const float* Wq = (const float*)d_in[4];  const float* bq = (const float*)d_in[5];
    const float* Wk = (const float*)d_in[6];  const float* bk = (const float*)d_in[7];
    const float* Wo = (const float*)d_in[8];  const float* bo = (const float*)d_in[9];
    const float* W1 = (const float*)d_in[10]; const float* b1 = (const float*)d_in[11];
    const float* W2 = (const float*)d_in[12]; const float* b2 = (const float*)d_in[13];
    const float* ln1_s = (const float*)d_in[14]; const float* ln1_b = (const float*)d_in[15];
    const float* ln2_s = (const float*)d_in[16]; const float* ln2_b = (const float*)d_in[17];

    float* out_h = (float*)d_out;
    float* newv  = out_h + E_DIM;                    // new_cache[0]
    float* newk  = newv + (size_t)S_LEN * E_DIM;     // new_cache[1]
    const float* oldv = cache;
    const float* oldk = cache + (size_t)S_LEN * E_DIM;

    float* ws       = (float*)d_ws;
    float* q        = ws;                        // 1024
    float* scores   = q + E_DIM;                 // 16*8192 (probs in-place)
    float* values   = scores + H_N * S_LEN;      // 1024
    float* h1       = values + E_DIM;            // 1024
    float* part_qkv = h1 + E_DIM;                // 3*4*1024
    float* part_o   = part_qkv + 12 * E_DIM;     // 8*1024
    float* part_1   = part_o + 8 * E_DIM;        // 8*1024
    float* part_2   = part_1 + 8 * E_DIM;        // 8*1024

    qkv_part_kernel<<<dim3(8, 4, 3), 256, 0, stream>>>(x, Wv, Wq, Wk, part_qkv);
    qkv_combine_kernel<<<1, 256, 0, stream>>>(part_qkv, bv, bq, bk, q, newv, newk);
    shift_score_kernel<<<256, 256, 0, stream>>>(oldv, oldk, q, newv, newk, scores);
    softmax_kernel<<<16, 256, 0, stream>>>(scores, values);
    attnv_kernel<<<128, 256, 0, stream>>>(scores, newv, values);
    gemv_part_kernel<<<dim3(8, 8), 256, 0, stream>>>(Wo, values, 1, nullptr, 0, part_o);
    ln_combine_kernel<<<1, 256, 0, stream>>>(part_o, 8, bo, x, ln1_s, ln1_b, h1);
    gemv_part_kernel<<<dim3(8, 8), 256, 0, stream>>>(W1, h1, 1, nullptr, 0, part_1);
    gemv_part_kernel<<<dim3(8, 8), 256, 0, stream>>>(W2, part_1, 8, b1, 1, part_2);
    ln_combine_kernel<<<1, 256, 0, stream>>>(part_2, 8, b2, h1, ln2_s, ln2_b, out_h);
}